// RNNEncoder_68539088109906
// MI455X (gfx1250) — compile-verified
//
#include <hip/hip_runtime.h>

// ---------------------------------------------------------------------------
// Bidirectional 2-layer LSTM encoder for MI455X (gfx1250, wave32, WMMA).
// B=32, S=1024, I=256, H=256, gates 4H=1024.
//
// proj_gemm: G[s*B+b,:] = x[s,b,:] @ Wih^T + (bih+bhh).  f32 WMMA 16x16x4,
//            one 16x64 strip per wave (4 N-tiles share each A fragment).
// lstm_rec:  persistent 1-workgroup-per-direction recurrence. h in padded
//            LDS, c in accumulator-layout VGPRs, i/f/g/o tiles per wave so
//            gate math runs on WMMA accumulators. G[t+1] is double-buffered
//            into LDS with GLOBAL_LOAD_ASYNC_TO_LDS_B128 (ASYNCcnt) so the
//            seed loads come off the sequential critical path.
// ---------------------------------------------------------------------------

typedef __attribute__((ext_vector_type(2))) float v2f;
typedef __attribute__((ext_vector_type(8))) float v8f;

#define BATCH 32
#define HID   256
#define GATES 1024      // 4*HID
#define SEQ   1024
#define HSTR  257       // padded LDS row stride (floats) for h buffer
#define LDS_SPACE __attribute__((address_space(3)))

__device__ __forceinline__ v8f wmma_f32_k4(v2f a, v2f b, v8f c) {
    // D = A(16x4 f32) * B(4x16 f32) + C(16x16 f32)
    return __builtin_amdgcn_wmma_f32_16x16x4_f32(
        false, a, false, b, (short)0, c, false, false);
}

__device__ __forceinline__ float sigf(float x) {
    return 1.0f / (1.0f + __expf(-x));
}

// Async global -> LDS copy, 16B per lane, tracked by ASYNCcnt.
__device__ __forceinline__ void async_g2l_b128(float* lds_dst,
                                               const float* gsrc) {
    void LDS_SPACE* ldst = (void LDS_SPACE*)lds_dst;
    asm volatile("global_load_async_to_lds_b128 %0, %1, off"
                 :: "v"(ldst), "v"(gsrc)
                 : "memory");
}

__device__ __forceinline__ void wait_async0() {
    asm volatile("s_wait_asynccnt 0" ::: "memory");
}

// ---------------------------------------------------------------------------
// Input projection: one 16x64 strip per wave (4 accumulators, shared A),
// 8 waves per block.  X element (row = s*B+b, k) at X[s*strideS+b*strideB+k].
// W is [GATES, K] row-major.  blockIdx.y selects fwd(0)/bwd(1) params.
// ---------------------------------------------------------------------------
__global__ __launch_bounds__(256) void proj_gemm(
    const float* __restrict__ X, long strideS, long strideB, int K,
    const float* __restrict__ Wf, const float* __restrict__ bihf,
    const float* __restrict__ bhhf, float* __restrict__ Gf,
    const float* __restrict__ Wb, const float* __restrict__ bihb,
    const float* __restrict__ bhhb, float* __restrict__ Gb)
{
    const int lane = threadIdx.x & 31;
    const int wave = threadIdx.x >> 5;

    const float* W;  const float* bih; const float* bhh; float* G;
    if (blockIdx.y == 0) { W = Wf; bih = bihf; bhh = bhhf; G = Gf; }
    else                 { W = Wb; bih = bihb; bhh = bhhb; G = Gb; }

    const int NSTRIP = GATES / 64;               // 16 N-strips of width 64
    const int strip  = blockIdx.x * 8 + wave;    // 2048*16 = 32768 strips
    const int ns = strip % NSTRIP;
    const int mt = strip / NSTRIP;
    const int m0 = mt * 16;
    const int n0 = ns * 64;

    const int l15  = lane & 15;
    const int half = lane >> 4;                  // 0: K={k,k+1}, 1: K={k+2,k+3}

    // A fragment source: row m0+l15 of X, starting at k offset 2*half
    const int arow = m0 + l15;                   // == s*B + b
    const int s = arow >> 5;
    const int b = arow & 31;
    const float* Arow = X + (long)s * strideS + (long)b * strideB + 2 * half;

    // B fragment sources: rows (n0 + j*16 + l15) of W, j = 0..3
    const int bcol = n0 + l15;
    const float* Wr0 = W + (long)(bcol +  0) * K + 2 * half;
    const float* Wr1 = W + (long)(bcol + 16) * K + 2 * half;
    const float* Wr2 = W + (long)(bcol + 32) * K + 2 * half;
    const float* Wr3 = W + (long)(bcol + 48) * K + 2 * half;

    v8f a0 = {}, a1 = {}, a2 = {}, a3 = {};
    for (int k = 0; k < K; k += 4) {
        v2f a, b0, b1, b2, b3;
        a.x  = Arow[k]; a.y  = Arow[k + 1];
        b0.x = Wr0[k];  b0.y = Wr0[k + 1];
        b1.x = Wr1[k];  b1.y = Wr1[k + 1];
        b2.x = Wr2[k];  b2.y = Wr2[k + 1];
        b3.x = Wr3[k];  b3.y = Wr3[k + 1];
        a0 = wmma_f32_k4(a, b0, a0);
        a1 = wmma_f32_k4(a, b1, a1);
        a2 = wmma_f32_k4(a, b2, a2);
        a3 = wmma_f32_k4(a, b3, a3);
    }

    const int rbase = m0 + (half << 3);          // D rows: r + 8*half
    const float bv0 = bih[bcol +  0] + bhh[bcol +  0];
    const float bv1 = bih[bcol + 16] + bhh[bcol + 16];
    const float bv2 = bih[bcol + 32] + bhh[bcol + 32];
    const float bv3 = bih[bcol + 48] + bhh[bcol + 48];
#pragma unroll
    for (int r = 0; r < 8; ++r) {
        float* Grow = G + (long)(rbase + r) * GATES + bcol;
        Grow[ 0] = a0[r] + bv0;
        Grow[16] = a1[r] + bv1;
        Grow[32] = a2[r] + bv2;
        Grow[48] = a3[r] + bv3;
    }
}

// ---------------------------------------------------------------------------
// Recurrence: grid.x = 2 (dir), block = 1024 threads = 32 waves on one WGP.
// Wave w: batch-tile (w>>4), hidden-tile (w&15); computes i,f,g,o 16x16
// tiles for its patch each step.
// ---------------------------------------------------------------------------
__global__ __launch_bounds__(1024) void lstm_rec(
    const float* __restrict__ Gfwd, const float* __restrict__ Gbwd,
    const float* __restrict__ Whf,  const float* __restrict__ Whb,
    const float* __restrict__ mask,                 // [BATCH, SEQ]
    float* __restrict__ xout, long outStrideT, long outStrideB,
    float* __restrict__ hnOut, float* __restrict__ cnOut)   // [BATCH, 2H]
{
    __shared__ float hbuf[BATCH * HSTR];            // ~33 KB
    __shared__ float gbuf[2][BATCH * GATES];        // 2 x 128 KB staged gates

    const int dir = blockIdx.x;                     // 0 = fwd, 1 = bwd
    const float* G  = dir ? Gbwd : Gfwd;
    const float* Wh = dir ? Whb  : Whf;             // [GATES, HID] row-major

    const int lane  = threadIdx.x & 31;
    const int wave  = threadIdx.x >> 5;
    const int mtile = wave >> 4;                    // 0..1 (batch tiles)
    const int jt    = wave & 15;                    // 0..15 (hidden tiles)
    const int l15   = lane & 15;
    const int half  = lane >> 4;

    const int col   = jt * 16 + l15;                // hidden index j
    const int rbase = mtile * 16 + (half << 3);     // D-layout batch row base
    const int arowb = mtile * 16 + l15;             // A-layout batch row

    // B-fragment base pointers for the four gates (rows of Whh)
    const float* Wi = Wh + (long)(0 * HID + col) * HID + 2 * half;
    const float* Wf = Wh + (long)(1 * HID + col) * HID + 2 * half;
    const float* Wg = Wh + (long)(2 * HID + col) * HID + 2 * half;
    const float* Wo = Wh + (long)(3 * HID + col) * HID + 2 * half;
    const float* hrow = &hbuf[arowb * HSTR + 2 * half];

    // h := 0
    for (int i = threadIdx.x; i < BATCH * HSTR; i += 1024) hbuf[i] = 0.0f;
    float creg[8];
#pragma unroll
    for (int r = 0; r < 8; ++r) creg[r] = 0.0f;

    // Stage G for the first step into gbuf[0] (async, 16B/lane x 8)
    {
        const int t0 = dir ? (SEQ - 1) : 0;
        const float* gsrc = G + (long)t0 * (BATCH * GATES);
        const int base = threadIdx.x * 4;
#pragma unroll
        for (int u = 0; u < 8; ++u) {
            const int off = u * 4096 + base;
            async_g2l_b128(&gbuf[0][off], gsrc + off);
        }
    }

    for (int step = 0; step < SEQ; ++step) {
        const int t   = dir ? (SEQ - 1 - step) : step;
        const int cur = step & 1;

        wait_async0();              // our async loads into gbuf[cur] done
        __syncthreads();            // everyone's loads + prior h-writes visible

        // seed accumulators from staged input gates (bias folded in)
        const float* Grow = &gbuf[cur][(rbase) * GATES + col];
        v8f ai, af, ag, ao;
#pragma unroll
        for (int r = 0; r < 8; ++r) {
            ai[r] = Grow[r * GATES + 0 * HID];
            af[r] = Grow[r * GATES + 1 * HID];
            ag[r] = Grow[r * GATES + 2 * HID];
            ao[r] = Grow[r * GATES + 3 * HID];
        }

        // kick off async staging of the next step's gates
        {
            const int tn = (step + 1 < SEQ) ? (dir ? (SEQ - 2 - step)
                                                   : (step + 1)) : t;
            const float* gsrc = G + (long)tn * (BATCH * GATES);
            const int base = threadIdx.x * 4;
#pragma unroll
            for (int u = 0; u < 8; ++u) {
                const int off = u * 4096 + base;
                async_g2l_b128(&gbuf[cur ^ 1][off], gsrc + off);
            }
        }

        // gates += h @ Whh^T  (K = HID = 256, 4 f32 WMMAs per k-step)
#pragma unroll 4
        for (int k = 0; k < HID; k += 4) {
            v2f a, bi, bf, bg, bo;
            a.x  = hrow[k]; a.y  = hrow[k + 1];
            bi.x = Wi[k];   bi.y = Wi[k + 1];
            bf.x = Wf[k];   bf.y = Wf[k + 1];
            bg.x = Wg[k];   bg.y = Wg[k + 1];
            bo.x = Wo[k];   bo.y = Wo[k + 1];
            ai = wmma_f32_k4(a, bi, ai);
            af = wmma_f32_k4(a, bf, af);
            ag = wmma_f32_k4(a, bg, ag);
            ao = wmma_f32_k4(a, bo, ao);
        }

        // LSTM cell update directly on accumulator registers
        float hvals[8];
#pragma unroll
        for (int r = 0; r < 8; ++r) {
            const int brow = rbase + r;
            const float mval = mask[(long)brow * SEQ + t];
            const float iv = sigf(ai[r]);
            const float fv = sigf(af[r]);
            const float gv = tanhf(ag[r]);
            const float ov = sigf(ao[r]);
            float cnv = fv * creg[r] + iv * gv;
            float hnv = ov * tanhf(cnv);
            hnv *= mval;            // h0 == 0 -> masked reset is scaling
            cnv *= mval;            // c0 == 0
            creg[r]  = cnv;
            hvals[r] = hnv;
            xout[(long)t * outStrideT + (long)brow * outStrideB
                 + dir * HID + col] = hnv;
        }

        __syncthreads();            // all reads of old h done
#pragma unroll
        for (int r = 0; r < 8; ++r)
            hbuf[(rbase + r) * HSTR + col] = hvals[r];
    }

    __syncthreads();
    // final states
#pragma unroll
    for (int r = 0; r < 8; ++r) {
        const int brow = rbase + r;
        hnOut[(long)brow * (2 * HID) + dir * HID + col] =
            hbuf[brow * HSTR + col];
        cnOut[(long)brow * (2 * HID) + dir * HID + col] = creg[r];
    }
}

// ---------------------------------------------------------------------------
// Host launcher
// ---------------------------------------------------------------------------
extern "C" void kernel_launch(void* const* d_in, const int* in_sizes, int n_in,
                              void* d_out, int out_size, void* d_ws, size_t ws_size,
                              hipStream_t stream) {
    const float* inputs = (const float*)d_in[0];   // [B, S, I]
    const float* mask   = (const float*)d_in[1];   // [B, S]
    const float* l0f_Wih = (const float*)d_in[2];
    const float* l0f_Whh = (const float*)d_in[3];
    const float* l0f_bih = (const float*)d_in[4];
    const float* l0f_bhh = (const float*)d_in[5];
    const float* l0b_Wih = (const float*)d_in[6];
    const float* l0b_Whh = (const float*)d_in[7];
    const float* l0b_bih = (const float*)d_in[8];
    const float* l0b_bhh = (const float*)d_in[9];
    const float* l1f_Wih = (const float*)d_in[10];
    const float* l1f_Whh = (const float*)d_in[11];
    const float* l1f_bih = (const float*)d_in[12];
    const float* l1f_bhh = (const float*)d_in[13];
    const float* l1b_Wih = (const float*)d_in[14];
    const float* l1b_Whh = (const float*)d_in[15];
    const float* l1b_bih = (const float*)d_in[16];
    const float* l1b_bhh = (const float*)d_in[17];

    // Workspace: Gf (134MB) | Gb (134MB) | x1 (67MB)
    float* ws = (float*)d_ws;
    const long GSZ = (long)SEQ * BATCH * GATES;    // 33,554,432 floats
    float* Gf = ws;
    float* Gb = ws + GSZ;
    float* x1 = ws + 2 * GSZ;                      // [S, B, 512]

    float* out    = (float*)d_out;                 // output [B, S, 512]
    float* hnBase = out + (long)BATCH * SEQ * (2 * HID);
    float* cnBase = hnBase + 2L * BATCH * (2 * HID);
    const long LAYSZ = (long)BATCH * (2 * HID);    // per-layer hn/cn slot

    const dim3 pb(256);
    const dim3 pg((2048 * 16) / 8, 2);             // 32768 strips, 8 waves/blk

    // ---- layer 0 ----
    proj_gemm<<<pg, pb, 0, stream>>>(
        inputs, /*strideS=*/256, /*strideB=*/(long)SEQ * 256, /*K=*/256,
        l0f_Wih, l0f_bih, l0f_bhh, Gf,
        l0b_Wih, l0b_bih, l0b_bhh, Gb);
    lstm_rec<<<dim3(2), dim3(1024), 0, stream>>>(
        Gf, Gb, l0f_Whh, l0b_Whh, mask,
        x1, /*outStrideT=*/(long)BATCH * 512, /*outStrideB=*/512,
        hnBase + 0 * LAYSZ, cnBase + 0 * LAYSZ);

    // ---- layer 1 ----
    proj_gemm<<<pg, pb, 0, stream>>>(
        x1, /*strideS=*/(long)BATCH * 512, /*strideB=*/512, /*K=*/512,
        l1f_Wih, l1f_bih, l1f_bhh, Gf,
        l1b_Wih, l1b_bih, l1b_bhh, Gb);
    lstm_rec<<<dim3(2), dim3(1024), 0, stream>>>(
        Gf, Gb, l1f_Whh, l1b_Whh, mask,
        out, /*outStrideT=*/512, /*outStrideB=*/(long)SEQ * 512,
        hnBase + 1 * LAYSZ, cnBase + 1 * LAYSZ);
}